// AttFusion_vit_90159953477995
// MI455X (gfx1250) — compile-verified
//
#include <hip/hip_runtime.h>
#include <hip/hip_bf16.h>

typedef __attribute__((ext_vector_type(2))) float v2f;
typedef __attribute__((ext_vector_type(8))) float v8f;

#define C_DIM    256
#define D_DIM    16384
#define TILE     2048
#define PAD      16            // LDS row pad (dwords) -> bank-conflict-free A loads
#define ROWS     4             // LMAX
#define NTHREADS 256

__device__ __forceinline__ unsigned lds_byte_off(const void* p) {
  // Generic pointers to LDS carry the AS(3) offset in the low 32 bits.
  return (unsigned)(unsigned long long)p;
}

__global__ __launch_bounds__(NTHREADS) void attfusion_kernel(
    const float* __restrict__ x,
    const int*   __restrict__ rl,
    int    n_groups_max,
    unsigned long long out_total,
    float* __restrict__ out)
{
  __shared__ float tileS[ROWS * (TILE + PAD)];  // ~33 KB ctx staging
  __shared__ float redS[8][16];                 // per-wave Gram partials
  __shared__ float attnS[ROWS][ROWS];           // softmax weights

  const int c      = blockIdx.x;
  const int g      = blockIdx.y;
  const int tid    = threadIdx.x;
  const int lane   = tid & 31;
  const int wave   = tid >> 5;
  const int lane15 = lane & 15;
  const int half   = lane >> 4;
  const int kbase  = half * 2;

  // Exact output-bound guard (protects against group-count ambiguity).
  if ((unsigned long long)(g + 1) * C_DIM * D_DIM > out_total) return;

  // record_len may arrive as i32 or i64; true lens are >= 1, so an i64
  // buffer viewed as i32 has rl[1] == 0 (high word of lens[0]).
  const bool is64 = (n_groups_max > 1) && (rl[0] != 0) && (rl[1] == 0);
  long long start = 0, L = 0;
  for (int i = 0; i <= g; ++i) {
    long long li = is64 ? ((const long long*)rl)[i] : (long long)rl[i];
    if (i < g) start += li; else L = li;
  }
  if (L < 1 || L > ROWS) return;          // workload guarantee: L in 1..4
  const int Li = (int)L;
  if (c * Li >= C_DIM) return;            // channel contributes no output
  const int nq = (Li < C_DIM - c * Li) ? Li : (C_DIM - c * Li);

  const float* base = x + ((size_t)start * C_DIM + (size_t)c * Li) * (size_t)D_DIM;

  // ---------------- Pass 1: Gram matrix G = V V^T (fp32, one HBM read) -----
  float acc[ROWS][ROWS];
  #pragma unroll
  for (int i = 0; i < ROWS; ++i)
    #pragma unroll
    for (int j = 0; j < ROWS; ++j) acc[i][j] = 0.f;

  for (int t = tid * 4; t < D_DIM; t += NTHREADS * 4) {
    float4 v[ROWS];
    for (int k = 0; k < Li; ++k)
      v[k] = *(const float4*)(base + (size_t)k * D_DIM + t);
    for (int i = 0; i < Li; ++i)
      for (int j = i; j < Li; ++j) {
        float a = acc[i][j];
        a = fmaf(v[i].x, v[j].x, a);
        a = fmaf(v[i].y, v[j].y, a);
        a = fmaf(v[i].z, v[j].z, a);
        a = fmaf(v[i].w, v[j].w, a);
        acc[i][j] = a;
      }
  }
  for (int i = 0; i < Li; ++i)
    for (int j = i; j < Li; ++j) {
      float r = acc[i][j];
      for (int o = 16; o > 0; o >>= 1) r += __shfl_xor(r, o, 32);
      if (lane == 0) redS[wave][i * ROWS + j] = r;
    }
  __syncthreads();

  if (tid == 0) {
    float G[ROWS][ROWS];
    for (int i = 0; i < Li; ++i)
      for (int j = i; j < Li; ++j) {
        float s = 0.f;
        for (int w = 0; w < 8; ++w) s += redS[w][i * ROWS + j];
        G[i][j] = s; G[j][i] = s;
      }
    const float scale = 0.0625f;  // 1/sqrt(FEATURE_DIM=256)
    for (int q = 0; q < nq; ++q) {
      float m = -3.402823466e38f;
      for (int k = 0; k < Li; ++k) m = fmaxf(m, G[q][k] * scale);
      float e[ROWS], sum = 0.f;
      for (int k = 0; k < Li; ++k) { e[k] = expf(G[q][k] * scale - m); sum += e[k]; }
      float inv = 1.f / sum;
      for (int k = 0; k < Li; ++k) attnS[q][k] = e[k] * inv;
    }
  }
  __syncthreads();

  // B fragment = attn^T for V_WMMA_F32_16X16X4_F32 (built once).
  // B 4x16 f32 layout: VGPR0 = {K0 | K2}, VGPR1 = {K1 | K3} by lane half,
  // column N = lane&15. Zero-pad k>=L and q>=nq.
  v2f Bfrag;
  {
    const int qc = lane15 & (ROWS - 1);   // clamp for safe LDS indexing
    float b0 = attnS[qc][kbase + 0];
    float b1 = attnS[qc][kbase + 1];
    Bfrag.x = (lane15 < nq && kbase + 0 < Li) ? b0 : 0.f;
    Bfrag.y = (lane15 < nq && kbase + 1 < Li) ? b1 : 0.f;
  }

  // Zero-fill unused tile rows once (async never writes rows k>=Li), so the
  // per-chunk A loads need no guards and stale-NaN 0*NaN poisoning is gone.
  for (int k = Li; k < ROWS; ++k)
    for (int pos = tid * 4; pos < TILE; pos += NTHREADS * 4)
      *(float4*)&tileS[k * (TILE + PAD) + pos] = make_float4(0.f, 0.f, 0.f, 0.f);

  const unsigned tileBase = lds_byte_off(tileS);
  const size_t   outBase  = ((size_t)g * C_DIM + (size_t)c * Li) * (size_t)D_DIM;
  const bool     do_store = lane15 < nq;  // loop-invariant store predicate

  // ---------------- Pass 2: ctx = V x attn^T via WMMA, async-staged --------
  for (int tb = 0; tb < D_DIM; tb += TILE) {
    // Async copy Li x TILE floats into padded LDS rows.
    for (int k = 0; k < Li; ++k)
      for (int pos = tid * 4; pos < TILE; pos += NTHREADS * 4) {
        unsigned long long gaddr =
            (unsigned long long)(base + (size_t)k * D_DIM + tb + pos);
        unsigned laddr = tileBase + (unsigned)((k * (TILE + PAD) + pos) * 4);
        asm volatile("global_load_async_to_lds_b128 %0, %1, off"
                     :: "v"(laddr), "v"(gaddr) : "memory");
      }
    asm volatile("s_wait_asynccnt 0x0" ::: "memory");
    __syncthreads();

    // Each wave owns 16-column chunks; one WMMA per chunk (K=4 >= L).
    // A[M=t_off][K=k] = v_k[t0+t_off]: lanes 0-15 M=lane (K0/K1),
    // lanes 16-31 M=lane-16 (K2/K3) -> same dual LDS fetch as before.
    #pragma unroll 2
    for (int ch = wave; ch < TILE / 16; ch += 8) {
      const int toff = ch * 16;
      v2f Af;
      Af.x = tileS[(kbase + 0) * (TILE + PAD) + toff + lane15];
      Af.y = tileS[(kbase + 1) * (TILE + PAD) + toff + lane15];
      v8f Cz = {0.f, 0.f, 0.f, 0.f, 0.f, 0.f, 0.f, 0.f};
      v8f Dm = __builtin_amdgcn_wmma_f32_16x16x4_f32(
          false, Af, false, Bfrag, (short)0, Cz, false, false);
      // D[M=t_off][N=q]: lane holds col q=lane15, VGPR j -> t_off j+8*half.
      // Dm[0..7] are 8 consecutive t values -> two b128 stores per lane.
      if (do_store) {
        size_t o = outBase + (size_t)lane15 * D_DIM
                 + (size_t)(tb + toff + 8 * half);
        float4 lo = make_float4(Dm[0], Dm[1], Dm[2], Dm[3]);
        float4 hi = make_float4(Dm[4], Dm[5], Dm[6], Dm[7]);
        *(float4*)(out + o)     = lo;
        *(float4*)(out + o + 4) = hi;
      }
    }
    __syncthreads();
  }
}

extern "C" void kernel_launch(void* const* d_in, const int* in_sizes, int n_in,
                              void* d_out, int out_size, void* d_ws, size_t ws_size,
                              hipStream_t stream) {
  const float* x  = (const float*)d_in[0];
  const int*   rl = (const int*)d_in[1];
  int n_groups = (n_in > 1) ? in_sizes[1] : 1;
  if (n_groups < 1) n_groups = 1;
  float* out = (float*)d_out;

  dim3 grid(C_DIM, n_groups);
  dim3 block(NTHREADS);
  attfusion_kernel<<<grid, block, 0, stream>>>(
      x, rl, n_groups, (unsigned long long)out_size, out);
}